// PositionalEncoder_1365799600364
// MI455X (gfx1250) — compile-verified
//
#include <hip/hip_runtime.h>

#define D_MODEL 1024
#define SEQ_LEN 2048

typedef __attribute__((ext_vector_type(4))) float v4f;

// log2(10000)
#define LOG2_N 13.287712379549449f

// out[b, pos, d] = x[b, pos, d] + pe[pos, d]
//   pe[pos, 2k]   = sin(pos / N^(2*(2k)/D))
//   pe[pos, 2k+1] = cos(pos / N^(2*(2k+1)/D))    (faithful to reference's exponents)
//
// Each thread owns one float4 (4 consecutive dims, starting at an even dim) of one
// row, computes its 4 pe values once in registers, then streams all `batch` copies
// with 128-bit non-temporal loads/stores. Pure HBM-bound: ~268 MB traffic.
__global__ __launch_bounds__(256) void pe_add_kernel(const float* __restrict__ x,
                                                     float* __restrict__ out,
                                                     int batch) {
    const int D4 = D_MODEL / 4; // 256 float4 per row
    const int idx = blockIdx.x * blockDim.x + threadIdx.x; // [0, SEQ_LEN*D4)
    if (idx >= SEQ_LEN * D4) return;

    const int pos = idx / D4;
    const int q   = idx - pos * D4;
    const int d0  = q * 4; // even

    const float pos_f = (float)pos;
    const float inv_d = 2.0f / (float)D_MODEL;

    // divisor(i) = N^(2*i/D) = exp2(log2(N) * 2*i/D)
    v4f pe;
    pe.x = __builtin_sinf(pos_f / __builtin_exp2f(LOG2_N * (inv_d * (float)(d0 + 0))));
    pe.y = __builtin_cosf(pos_f / __builtin_exp2f(LOG2_N * (inv_d * (float)(d0 + 1))));
    pe.z = __builtin_sinf(pos_f / __builtin_exp2f(LOG2_N * (inv_d * (float)(d0 + 2))));
    pe.w = __builtin_cosf(pos_f / __builtin_exp2f(LOG2_N * (inv_d * (float)(d0 + 3))));

    const size_t row_off      = (size_t)pos * D_MODEL + (size_t)d0;
    const size_t batch_stride = (size_t)SEQ_LEN * (size_t)D_MODEL;

    if (batch == 16) {
        // Fast path: fully unrolled -> 16 independent b128 NT loads in flight.
#pragma unroll
        for (int b = 0; b < 16; ++b) {
            const size_t off = (size_t)b * batch_stride + row_off;
            v4f v = __builtin_nontemporal_load((const v4f*)(x + off));
            v += pe;
            __builtin_nontemporal_store(v, (v4f*)(out + off));
        }
    } else {
#pragma unroll 4
        for (int b = 0; b < batch; ++b) {
            const size_t off = (size_t)b * batch_stride + row_off;
            v4f v = __builtin_nontemporal_load((const v4f*)(x + off));
            v += pe;
            __builtin_nontemporal_store(v, (v4f*)(out + off));
        }
    }
}

extern "C" void kernel_launch(void* const* d_in, const int* in_sizes, int n_in,
                              void* d_out, int out_size, void* d_ws, size_t ws_size,
                              hipStream_t stream) {
    (void)n_in; (void)out_size; (void)d_ws; (void)ws_size;

    const float* x  = (const float*)d_in[0];
    float* out      = (float*)d_out;
    const int batch = in_sizes[0] / (SEQ_LEN * D_MODEL); // 16 for the reference setup

    const int total_f4 = SEQ_LEN * (D_MODEL / 4); // 524,288 threads
    const int block = 256;                        // 8 wave32 waves
    const int grid  = (total_f4 + block - 1) / block; // 2048 blocks

    pe_add_kernel<<<grid, block, 0, stream>>>(x, out, batch);
}